// TransferMatrixMethod_28020366639147
// MI455X (gfx1250) — compile-verified
//
#include <hip/hip_runtime.h>
#include <hip/hip_bf16.h>
#include <cstdint>

#ifndef TMM_TWO_PI
#define TMM_TWO_PI 6.28318530717958647692f
#endif

// Transfer Matrix Method. The chain M_total = Prod_l [[cos,-i sin/n],[-i n sin,cos]]
// stays in the closed form [[A, iB],[iC, D]] with A,B,C,D real (n is real), so each
// layer is 8 real FMAs + one sincos. Per-layer constants (n, n*d, 1/(n+1e-8)) are
// wavelength-independent: computed once per block and packed into an LDS float4
// table (single ds_load_b128 broadcast per layer in the hot loop).
// Phase is fed to the TRANS units in *revolutions* (n*d/lambda), so the 2*pi
// multiply and the hardware 1/(2*pi) normalization cancel: one v_mul + v_sin +
// v_cos per layer. Layer rows are staged global->LDS via gfx1250 async DMA.
__global__ __launch_bounds__(256) void TransferMatrixMethod_28020366639147_kernel(
    const float* __restrict__ n_layers,  // (B, L)
    const float* __restrict__ d_layers,  // (B, L)
    const float* __restrict__ lam,       // (W,)
    float* __restrict__ out,             // (B, W)
    int L, int W)
{
    __shared__ float  s_n[256];
    __shared__ float  s_d[256];
    __shared__ float4 s_pack[256];   // {n, n*d, 1/(n+1e-8), unused}

    const int b = blockIdx.x;
    const int w = blockIdx.y * blockDim.x + threadIdx.x;

    const float* gn = n_layers + (size_t)b * (size_t)L;
    const float* gd = d_layers + (size_t)b * (size_t)L;

#if __has_builtin(__builtin_amdgcn_global_load_async_to_lds_b32)
    // CDNA5 async DMA: global -> LDS without touching VGPR data lanes.
    // Builtin signature: (int* AS1, int* AS3, imm offset, imm cpol)
    for (int i = threadIdx.x; i < L; i += blockDim.x) {
        __builtin_amdgcn_global_load_async_to_lds_b32(
            (__attribute__((address_space(1))) int*)(uintptr_t)(gn + i),
            (__attribute__((address_space(3))) int*)&s_n[i],
            0, 0);
        __builtin_amdgcn_global_load_async_to_lds_b32(
            (__attribute__((address_space(1))) int*)(uintptr_t)(gd + i),
            (__attribute__((address_space(3))) int*)&s_d[i],
            0, 0);
    }
#if __has_builtin(__builtin_amdgcn_s_wait_asynccnt)
    __builtin_amdgcn_s_wait_asynccnt(0);
#else
    asm volatile("s_wait_asynccnt 0" ::: "memory");
#endif
#else
    for (int i = threadIdx.x; i < L; i += blockDim.x) {
        s_n[i] = gn[i];
        s_d[i] = gd[i];
    }
#endif
    __syncthreads();

    // Hoist wavelength-independent per-layer work: one division per (b,layer)
    // instead of one per (b,layer,wavelength).
    for (int i = threadIdx.x; i < L; i += blockDim.x) {
        const float nv = s_n[i];
        const float dv = s_d[i];
        float4 p;
        p.x = nv;                      // n
        p.y = nv * dv;                 // n*d
        p.z = 1.0f / (nv + 1e-8f);     // 1/(n+1e-8)
        p.w = 0.0f;
        s_pack[i] = p;
    }
    __syncthreads();

    if (w >= W) return;

    const float inv_lam = 1.0f / lam[w];   // 1/lambda; phase in revolutions = n*d/lambda

    // M_total = [[A, i*Bm],[i*C, D]], start at identity.
    float A = 1.0f, Bm = 0.0f, C = 0.0f, D = 1.0f;

    const int Lm1 = L - 1;
#pragma unroll 4
    for (int l = 1; l < Lm1; ++l) {
        const float4 p = s_pack[l];        // single b128 LDS broadcast
        float s, c;
#if __has_builtin(__builtin_amdgcn_sinf) && __has_builtin(__builtin_amdgcn_cosf)
        const float t = p.y * inv_lam;     // revolutions; |t| <~ 1.9, in HW range
        s = __builtin_amdgcn_sinf(t);      // v_sin_f32: sin(2*pi*t)
        c = __builtin_amdgcn_cosf(t);      // v_cos_f32: cos(2*pi*t)
#else
        const float phi = p.y * (TMM_TWO_PI * inv_lam);
        __sincosf(phi, &s, &c);
#endif
        const float ns = p.x * s;          // n*sin(phi)
        const float sn = p.z * s;          // sin(phi)/(n+1e-8)

        const float nA = fmaf(Bm, ns, A * c);   // A' = A c + B n s
        const float nB = fmaf(-A, sn, Bm * c);  // B' = B c - A s/n
        const float nC = fmaf(-D, ns, C * c);   // C' = C c - D n s
        const float nD = fmaf(C, sn, D * c);    // D' = D c + C s/n
        A = nA; Bm = nB; C = nC; D = nD;
    }

    const float n_in  = s_n[0];
    const float n_sub = s_n[Lm1];

    // E = A + 1e-9 + i*(B*n_sub);  H = D*n_sub + i*C
    const float Er = A + 1e-9f;
    const float Ei = Bm * n_sub;
    const float Hr = D * n_sub;
    const float Hi = C;

    // Y = H/E (complex)
    const float inv_den = 1.0f / fmaf(Er, Er, Ei * Ei);
    const float Yr = fmaf(Hr, Er, Hi * Ei) * inv_den;
    const float Yi = fmaf(Hi, Er, -Hr * Ei) * inv_den;

    // r = (n_in - Y)/(n_in + Y);  R = |r|^2 = |num|^2 / |den|^2
    const float ar = n_in - Yr;
    const float br = n_in + Yr;
    const float num = fmaf(ar, ar, Yi * Yi);
    const float dnm = fmaf(br, br, Yi * Yi);

    out[(size_t)b * (size_t)W + (size_t)w] = num / dnm;
}

extern "C" void kernel_launch(void* const* d_in, const int* in_sizes, int n_inputs,
                              void* d_out, int out_size, void* d_ws, size_t ws_size,
                              hipStream_t stream) {
    (void)n_inputs; (void)d_ws; (void)ws_size;
    const float* n_layers = (const float*)d_in[0];
    const float* d_layers = (const float*)d_in[1];
    const float* lam      = (const float*)d_in[2];
    float* out = (float*)d_out;

    const int W = in_sizes[2];           // 512
    const int B = out_size / W;          // 256
    const int L = in_sizes[0] / B;       // 64

    dim3 grid((unsigned)B, (unsigned)((W + 255) / 256));
    TransferMatrixMethod_28020366639147_kernel<<<grid, 256, 0, stream>>>(
        n_layers, d_layers, lam, out, L, W);
}